// DVAE_INCI_WO_GNN_16045997818093
// MI455X (gfx1250) — compile-verified
//
#include <hip/hip_runtime.h>
#include <cstddef>

// ---------------------------------------------------------------------------
// DVAE encoder (inci graph VAE) for MI455X / gfx1250, wave32 + WMMA f32.
//
// hidden = emb[ops] has only NTYPE=10 distinct rows, so hidden@Wg ==
// (emb@Wg)[ops]; the two 10020x10020 Wg products drop from 308 GFLOP to
// 4 GFLOP and are purely HBM-bound (800MB @ 23.3 TB/s ~ 34us).
// The 12 inception GEMMs [1536x10020]x[10020x501] (~185 GFLOP fp32) run on
// V_WMMA_F32_16X16X4_F32, 64x64 tiles, 8 waves, 2 tiles/wave; weights are
// pre-padded to [10032x512] so the K-loop is branch-free; LDS is double-
// buffered (1 barrier/chunk, global loads overlap WMMA); B is stored pair-
// interleaved in LDS so fragments are single aligned b64 reads (no v_movs).
// mu/lv weight pairs share one launch via blockIdx.z for 2x blocks in flight.
// ---------------------------------------------------------------------------

#define B_    32
#define V_    24
#define E_    48
#define NT_   10
#define HID_  501
#define D_    10020
#define Z_    56
#define KP_   10032          // D_ padded to multiple of 16 (zero cols)
#define NP_   512            // HID_ padded to 512
#define M_    1536           // B_*E_
#define ROWS_ 120            // V_ + 2*E_
#define HK_   1536           // hcat K: 3 segments of 512 (501 real + 11 zero)
#define KSEG_ 627            // ceil(D_/16): K-split segment for emb@Wg
#define LN_EPS_ 1e-5f

typedef float v2f __attribute__((ext_vector_type(2)));
typedef float v8f __attribute__((ext_vector_type(8)));

// Assumed d_in order: top-level dict insertion order, params flattened with
// jax-style sorted keys at every nesting level (ASCII: capitals first).
enum {
  I_OPS = 0, I_INCI_T, I_INCI_H, I_NZ_V, I_NZ_ET, I_NZ_EH,
  I_WG_H, I_WG_T, I_WV_LV, I_WV_MU, I_BV_LV, I_BV_MU, I_EMB,
  I_LEH_B, I_LEH_G, I_LET_B, I_LET_G, I_LV_B, I_LV_G,
  I_MEH_B, I_MEH_G, I_MET_B, I_MET_G, I_MV_B, I_MV_G,
  I_LVH_W0, I_LVH_W1, I_LVH_W2, I_LVH_WOUT, I_LVH_BOUT,
  I_LVT_W0, I_LVT_W1, I_LVT_W2, I_LVT_WOUT, I_LVT_BOUT,
  I_MUH_W0, I_MUH_W1, I_MUH_W2, I_MUH_WOUT, I_MUH_BOUT,
  I_MUT_W0, I_MUT_W1, I_MUT_W2, I_MUT_WOUT, I_MUT_BOUT
};

// ---------------------------------------------------------------------------
// Paired fp32 WMMA GEMM: for z in {0,1}: C[z] = act(A[z] @ Wp[z] (+bias[z]))
// A: [M, lda] zero-padded beyond real K; Wp: fully padded [lda, ldw].
// Block = 256 thr (8 waves), tile 64x64, K-chunk 16 (4 wmma steps/chunk).
// Double-buffered LDS, one barrier per chunk, branch-free inner loop.
// Epilogue: cols in [Nw, Nwrite) written as 0 (pads the next GEMM's K).
// ---------------------------------------------------------------------------
__global__ __launch_bounds__(256)
void gemm_wmma_f32(const float* __restrict__ A0, const float* __restrict__ A1,
                   int lda,
                   const float* __restrict__ W0p, const float* __restrict__ W1p,
                   int ldw,
                   float* __restrict__ C0, float* __restrict__ C1, int ldc,
                   int Nw, int Nwrite,
                   const float* __restrict__ bias0, const float* __restrict__ bias1,
                   int do_relu)
{
  const float* A    = blockIdx.z ? A1 : A0;
  const float* W    = blockIdx.z ? W1p : W0p;
  float*       C    = blockIdx.z ? C1 : C0;
  const float* bias = blockIdx.z ? bias1 : bias0;

  // double-buffered tiles
  __shared__ float As[2][64][20];    // stride 20: conflict-free b64 frag reads
  __shared__ float Bs[2][8][160];    // [k-pair][2*col + (k&1)]; stride 160 ->
                                     // lane halves on disjoint banks, b64 frags

  const int tid  = threadIdx.x;
  const int lane = tid & 31;
  const int wave = tid >> 5;
  const int wm   = wave & 3;        // wave M tile (0..3)
  const int wn   = wave >> 2;       // wave N pair (0..1)
  const int m    = lane & 15;
  const int half = lane >> 4;
  const int koff = half * 2;        // ISA f32 A/B frag: lanes16-31 hold K+2

  const int blockM = blockIdx.y * 64;
  const int blockN = blockIdx.x * 64;

  v8f acc0 = {}; v8f acc1 = {};

  const int ar = tid >> 2;          // A loader: row 0..63
  const int ac = (tid & 3) * 4;     // 4 cols via float4
  const int br = tid >> 4;          // B loader: row 0..15
  const int bc = (tid & 15) * 4;
  const int bp = br >> 1;           // B k-pair
  const int bj = br & 1;            // slot within pair

  const float* aptr = A + (size_t)(blockM + ar) * lda + ac;
  const float* bptr = W + (size_t)br * ldw + blockN + bc;

  const int nchunks = lda >> 4;

  // prologue: stage chunk 0 into buffer 0
  float4 av = *(const float4*)(aptr);
  float4 bv = *(const float4*)(bptr);
  As[0][ar][ac + 0] = av.x; As[0][ar][ac + 1] = av.y;
  As[0][ar][ac + 2] = av.z; As[0][ar][ac + 3] = av.w;
  Bs[0][bp][2 * (bc + 0) + bj] = bv.x; Bs[0][bp][2 * (bc + 1) + bj] = bv.y;
  Bs[0][bp][2 * (bc + 2) + bj] = bv.z; Bs[0][bp][2 * (bc + 3) + bj] = bv.w;
  __syncthreads();

  for (int kc = 0; kc < nchunks; ++kc) {
    const int p  = kc & 1;
    const int q  = p ^ 1;
    const int kn = (kc + 1 < nchunks) ? (kc + 1) : kc;   // clamped: no branch

    // issue next chunk's global loads; latency hidden under the 8 WMMAs
    av = *(const float4*)(aptr + (size_t)kn * 16);
    bv = *(const float4*)(bptr + (size_t)kn * 16 * ldw);
    const int kn2 = (kc + 2 < nchunks) ? (kc + 2) : kn;
    __builtin_prefetch(aptr + (size_t)kn2 * 16, 0, 1);   // global_prefetch_b8

    const float* arow = &As[p][16 * wm + m][0];
    const float* brow = &Bs[p][0][0];
    const int cb2 = 2 * (32 * wn + m);
    #pragma unroll
    for (int kk = 0; kk < 16; kk += 4) {
      const int pr = (kk + koff) >> 1;                   // k-pair index
      v2f a  = *(const v2f*)(arow + kk + koff);          // {A[m][k0],A[m][k0+1]}
      v2f b0 = *(const v2f*)(brow + pr * 160 + cb2);     // {B[k0][n],B[k0+1][n]}
      v2f b1 = *(const v2f*)(brow + pr * 160 + cb2 + 32);
      acc0 = __builtin_amdgcn_wmma_f32_16x16x4_f32(false, a, false, b0,
                                                   (short)0, acc0, false, false);
      acc1 = __builtin_amdgcn_wmma_f32_16x16x4_f32(false, a, false, b1,
                                                   (short)0, acc1, false, false);
    }

    // stage next chunk into the other buffer (its last readers passed the
    // previous barrier, so one barrier per chunk is sufficient)
    As[q][ar][ac + 0] = av.x; As[q][ar][ac + 1] = av.y;
    As[q][ar][ac + 2] = av.z; As[q][ar][ac + 3] = av.w;
    Bs[q][bp][2 * (bc + 0) + bj] = bv.x; Bs[q][bp][2 * (bc + 1) + bj] = bv.y;
    Bs[q][bp][2 * (bc + 2) + bj] = bv.z; Bs[q][bp][2 * (bc + 3) + bj] = bv.w;
    __syncthreads();
  }

  // ---- epilogue: C VGPR r -> rows r / r+8 (lane halves), col = lane&15 ----
  #pragma unroll
  for (int r = 0; r < 8; ++r) {
    const int row = blockM + 16 * wm + r + half * 8;
    #pragma unroll
    for (int t = 0; t < 2; ++t) {
      const int col = blockN + 32 * wn + 16 * t + m;
      if (col < Nwrite) {
        float v = (t == 0) ? acc0[r] : acc1[r];
        if (col < Nw) {
          if (bias) v += bias[col];
          if (do_relu) v = fmaxf(v, 0.f);
        } else {
          v = 0.f;                    // zero-pad cols feed next GEMM's K
        }
        C[(size_t)row * ldc + col] = v;
      }
    }
  }
}

// Pad inception weight [D_, HID_] -> [KP_, NP_] (zeros outside).
__global__ void k_wpad(const float* __restrict__ W, float* __restrict__ Wp)
{
  const int i = blockIdx.x * 256 + threadIdx.x;
  if (i >= KP_ * NP_) return;
  const int kp = i >> 9, n = i & (NP_ - 1);
  float v = 0.f;
  if (kp < D_ && n < HID_) v = W[(size_t)kp * HID_ + n];
  Wp[i] = v;
}

// ---------------------------------------------------------------------------
// emb @ Wg partials: (10 x D) @ (D x D), K split 16 ways (no atomics ->
// deterministic). Streams the 400MB Wg coalesced; reduced by k_reduce16.
// ---------------------------------------------------------------------------
__global__ __launch_bounds__(256)
void k_embW_big(const float* __restrict__ emb, const float* __restrict__ Wg,
                float* __restrict__ part)
{
  const int col = blockIdx.x * 256 + threadIdx.x;
  if (col >= D_) return;
  const int k0 = blockIdx.y * KSEG_;
  const int k1 = (k0 + KSEG_ < D_) ? (k0 + KSEG_) : D_;
  float acc[NT_];
  #pragma unroll
  for (int t = 0; t < NT_; ++t) acc[t] = 0.f;
  for (int k = k0; k < k1; ++k) {
    const float w = Wg[(size_t)k * D_ + col];
    #pragma unroll
    for (int t = 0; t < NT_; ++t)
      acc[t] = fmaf(emb[(size_t)t * D_ + k], w, acc[t]);
  }
  float* dst = part + (size_t)blockIdx.y * (NT_ * D_);
  #pragma unroll
  for (int t = 0; t < NT_; ++t) dst[(size_t)t * D_ + col] = acc[t];
}

__global__ void k_reduce16(const float* __restrict__ part, float* __restrict__ out, int n)
{
  const int i = blockIdx.x * 256 + threadIdx.x;
  if (i >= n) return;
  float s = 0.f;
  #pragma unroll
  for (int j = 0; j < 16; ++j) s += part[(size_t)j * n + i];
  out[i] = s;
}

// emb @ Wv : (10 x D) @ (D x 56) -> (10 x 56). Tiny.
__global__ void k_embW_small(const float* __restrict__ emb, const float* __restrict__ Wv,
                             float* __restrict__ out)
{
  const int t = blockIdx.x;
  const int z = threadIdx.x;
  if (z >= Z_) return;
  float acc = 0.f;
  for (int k = 0; k < D_; ++k)
    acc = fmaf(emb[(size_t)t * D_ + k], Wv[(size_t)k * Z_ + z], acc);
  out[t * Z_ + z] = acc;
}

// edges[b,e,d] = relu( sum_v inci[b,v,e] * embWg[ops[b,v], d] ), ld=KP_, zero pad
__global__ __launch_bounds__(256)
void k_edges(const float* __restrict__ inci, const int* __restrict__ ops,
             const float* __restrict__ embWg, float* __restrict__ x)
{
  const int d = blockIdx.x * 256 + threadIdx.x;
  const int e = blockIdx.y, b = blockIdx.z;
  if (d >= KP_) return;
  float v = 0.f;
  if (d < D_) {
    #pragma unroll
    for (int vt = 0; vt < V_; ++vt) {
      const float w = inci[((size_t)b * V_ + vt) * E_ + e];
      const int op = ops[b * V_ + vt];
      v = fmaf(w, embWg[(size_t)op * D_ + d], v);
    }
    v = fmaxf(v, 0.f);
  }
  x[((size_t)b * E_ + e) * KP_ + d] = v;
}

// vx[b,v,d] = sum_e inci[b,v,e] * x[b,e,d]   (x ld=KP_, vx ld=D_)
__global__ __launch_bounds__(256)
void k_vx(const float* __restrict__ inci, const float* __restrict__ x,
          float* __restrict__ vx)
{
  const int d = blockIdx.x * 256 + threadIdx.x;
  const int v = blockIdx.y, b = blockIdx.z;
  if (d >= D_) return;
  float acc = 0.f;
  #pragma unroll 8
  for (int e = 0; e < E_; ++e)
    acc = fmaf(inci[((size_t)b * V_ + v) * E_ + e],
               x[((size_t)b * E_ + e) * KP_ + d], acc);
  vx[((size_t)b * V_ + v) * D_ + d] = acc;
}

// l[b,e,d] = sum_v inci[b,v,e] * vx[b,v,d]   (out ld=KP_, zero pad, no relu)
__global__ __launch_bounds__(256)
void k_lx(const float* __restrict__ inci, const float* __restrict__ vx,
          float* __restrict__ l)
{
  const int d = blockIdx.x * 256 + threadIdx.x;
  const int e = blockIdx.y, b = blockIdx.z;
  if (d >= KP_) return;
  float acc = 0.f;
  if (d < D_) {
    #pragma unroll
    for (int vt = 0; vt < V_; ++vt)
      acc = fmaf(inci[((size_t)b * V_ + vt) * E_ + e],
                 vx[((size_t)b * V_ + vt) * D_ + d], acc);
  }
  l[((size_t)b * E_ + e) * KP_ + d] = acc;
}

// Repack Wout[1503,56] into padded WoutP[1536,64] matching hcat segment layout.
__global__ void k_woutp(const float* __restrict__ Wout, float* __restrict__ Wp)
{
  const int i = blockIdx.x * 256 + threadIdx.x;
  if (i >= HK_ * 64) return;
  const int kp = i >> 6, n = i & 63;
  const int seg = kp >> 9, cc = kp & 511;
  float v = 0.f;
  if (cc < HID_ && n < Z_) v = Wout[((size_t)seg * HID_ + cc) * Z_ + n];
  Wp[i] = v;
}

__device__ __forceinline__ float blk_sum(float v, float* red)
{
  const int tid = threadIdx.x;
  red[tid] = v; __syncthreads();
  #pragma unroll
  for (int s = 128; s > 0; s >>= 1) {
    if (tid < s) red[tid] += red[tid + s];
    __syncthreads();
  }
  const float r = red[0]; __syncthreads();
  return r;
}

// Vertex path: relu(gathered emb@Wv + b), LayerNorm over (V,Z), reparameterize.
__global__ __launch_bounds__(256)
void k_vertex(const int* __restrict__ ops,
              const float* __restrict__ eWmu, const float* __restrict__ eWlv,
              const float* __restrict__ bmu,  const float* __restrict__ blv,
              const float* __restrict__ gm, const float* __restrict__ bm,
              const float* __restrict__ gl, const float* __restrict__ bl,
              const float* __restrict__ noise, float* __restrict__ out)
{
  __shared__ float smu[V_ * Z_], slv[V_ * Z_], red[256];
  const int b = blockIdx.x, tid = threadIdx.x;
  const int N = V_ * Z_;
  float s0 = 0, s1 = 0, s2 = 0, s3 = 0;
  for (int i = tid; i < N; i += 256) {
    const int v = i / Z_, z = i - v * Z_;
    const int op = ops[b * V_ + v];
    const float mu = fmaxf(eWmu[op * Z_ + z] + bmu[z], 0.f);
    const float lv = fmaxf(eWlv[op * Z_ + z] + blv[z], 0.f);
    smu[i] = mu; slv[i] = lv;
    s0 += mu; s1 += mu * mu; s2 += lv; s3 += lv * lv;
  }
  const float t0 = blk_sum(s0, red), t1 = blk_sum(s1, red);
  const float t2 = blk_sum(s2, red), t3 = blk_sum(s3, red);
  const float invN = 1.f / N;
  const float mm = t0 * invN, im = rsqrtf(t1 * invN - mm * mm + LN_EPS_);
  const float ml = t2 * invN, il = rsqrtf(t3 * invN - ml * ml + LN_EPS_);
  for (int i = tid; i < N; i += 256) {
    const float mun = (smu[i] - mm) * im * gm[i] + bm[i];
    const float lvn = (slv[i] - ml) * il * gl[i] + bl[i];
    out[(size_t)b * ROWS_ * Z_ + i] = noise[(size_t)b * N + i] * expf(0.5f * lvn) + mun;
  }
}

// Edge path: LayerNorm over (E,Z) for mean & logvar, reparameterize, scatter.
__global__ __launch_bounds__(256)
void k_edge_out(const float* __restrict__ mu_in, const float* __restrict__ lv_in,
                const float* __restrict__ gm, const float* __restrict__ bm,
                const float* __restrict__ gl, const float* __restrict__ bl,
                const float* __restrict__ noise, float* __restrict__ out, int rowbase)
{
  __shared__ float smu[E_ * Z_], slv[E_ * Z_], red[256];
  const int b = blockIdx.x, tid = threadIdx.x;
  const int N = E_ * Z_;
  float s0 = 0, s1 = 0, s2 = 0, s3 = 0;
  for (int i = tid; i < N; i += 256) {
    const float mu = mu_in[(size_t)b * N + i];
    const float lv = lv_in[(size_t)b * N + i];
    smu[i] = mu; slv[i] = lv;
    s0 += mu; s1 += mu * mu; s2 += lv; s3 += lv * lv;
  }
  const float t0 = blk_sum(s0, red), t1 = blk_sum(s1, red);
  const float t2 = blk_sum(s2, red), t3 = blk_sum(s3, red);
  const float invN = 1.f / N;
  const float mm = t0 * invN, im = rsqrtf(t1 * invN - mm * mm + LN_EPS_);
  const float ml = t2 * invN, il = rsqrtf(t3 * invN - ml * ml + LN_EPS_);
  for (int i = tid; i < N; i += 256) {
    const float mun = (smu[i] - mm) * im * gm[i] + bm[i];
    const float lvn = (slv[i] - ml) * il * gl[i] + bl[i];
    out[((size_t)b * ROWS_ + rowbase) * Z_ + i] =
        noise[(size_t)b * N + i] * expf(0.5f * lvn) + mun;
  }
}

// ---------------------------------------------------------------------------
extern "C" void kernel_launch(void* const* d_in, const int* in_sizes, int n_in,
                              void* d_out, int out_size, void* d_ws, size_t ws_size,
                              hipStream_t stream)
{
  (void)in_sizes; (void)n_in; (void)out_size; (void)ws_size;
  auto F = [&](int i) -> const float* { return (const float*)d_in[i]; };
  const int* ops = (const int*)d_in[I_OPS];
  float* ws  = (float*)d_ws;
  float* out = (float*)d_out;

  // workspace carve-out (floats, 64B aligned blocks) -- ~223 MB total
  size_t o = 0;
  auto alloc = [&](size_t n) { size_t r = o; o += (n + 15) & ~(size_t)15; return r; };
  float* part    = ws + alloc((size_t)16 * NT_ * D_);
  float* embWgT  = ws + alloc((size_t)NT_ * D_);
  float* embWgH  = ws + alloc((size_t)NT_ * D_);
  float* eWmu    = ws + alloc(NT_ * Z_);
  float* eWlv    = ws + alloc(NT_ * Z_);
  float* xbuf    = ws + alloc((size_t)M_ * KP_);   // edges, later l2
  float* l1buf   = ws + alloc((size_t)M_ * KP_);
  float* vxbuf   = ws + alloc((size_t)B_ * V_ * D_);
  float* hcat_mu = ws + alloc((size_t)M_ * HK_);
  float* hcat_lv = ws + alloc((size_t)M_ * HK_);
  float* WpA     = ws + alloc((size_t)KP_ * NP_);  // padded W (mu)
  float* WpB     = ws + alloc((size_t)KP_ * NP_);  // padded W (lv)
  float* WoM     = ws + alloc((size_t)HK_ * 64);   // padded Wout (mu)
  float* WoL     = ws + alloc((size_t)HK_ * 64);   // padded Wout (lv)
  float* outMu   = ws + alloc((size_t)M_ * Z_);
  float* outLv   = ws + alloc((size_t)M_ * Z_);

  const int nEmb = NT_ * D_;
  const int gWpad = (KP_ * NP_ + 255) / 256;

  // stage 0: emb@Wg (memory-bound, K-split partials), emb@Wv, vertex path
  k_embW_big<<<dim3(40, 16), 256, 0, stream>>>(F(I_EMB), F(I_WG_T), part);
  k_reduce16<<<dim3((nEmb + 255) / 256), 256, 0, stream>>>(part, embWgT, nEmb);
  k_embW_big<<<dim3(40, 16), 256, 0, stream>>>(F(I_EMB), F(I_WG_H), part);
  k_reduce16<<<dim3((nEmb + 255) / 256), 256, 0, stream>>>(part, embWgH, nEmb);
  k_embW_small<<<dim3(NT_), 64, 0, stream>>>(F(I_EMB), F(I_WV_MU), eWmu);
  k_embW_small<<<dim3(NT_), 64, 0, stream>>>(F(I_EMB), F(I_WV_LV), eWlv);
  k_vertex<<<dim3(B_), 256, 0, stream>>>(ops, eWmu, eWlv, F(I_BV_MU), F(I_BV_LV),
      F(I_MV_G), F(I_MV_B), F(I_LV_G), F(I_LV_B), F(I_NZ_V), out);

  // per-direction weight tables (X=0 -> T, X=1 -> H)
  const float* inciX[2]   = { F(I_INCI_T), F(I_INCI_H) };
  const float* embWgX[2]  = { embWgT, embWgH };
  const float* noiseX[2]  = { F(I_NZ_ET), F(I_NZ_EH) };
  const float* Wmu_[2][3] = { { F(I_MUT_W0), F(I_MUT_W1), F(I_MUT_W2) },
                              { F(I_MUH_W0), F(I_MUH_W1), F(I_MUH_W2) } };
  const float* Wlv_[2][3] = { { F(I_LVT_W0), F(I_LVT_W1), F(I_LVT_W2) },
                              { F(I_LVH_W0), F(I_LVH_W1), F(I_LVH_W2) } };
  const float* WoutMu[2] = { F(I_MUT_WOUT), F(I_MUH_WOUT) };
  const float* WoutLv[2] = { F(I_LVT_WOUT), F(I_LVH_WOUT) };
  const float* boutMu[2] = { F(I_MUT_BOUT), F(I_MUH_BOUT) };
  const float* boutLv[2] = { F(I_LVT_BOUT), F(I_LVH_BOUT) };
  const float* gme[2] = { F(I_MET_G), F(I_MEH_G) };
  const float* bme[2] = { F(I_MET_B), F(I_MEH_B) };
  const float* gle[2] = { F(I_LET_G), F(I_LEH_G) };
  const float* ble[2] = { F(I_LET_B), F(I_LEH_B) };
  const int rowbase[2] = { V_, V_ + E_ };

  // big GEMM + relu, mu & lv paired on blockIdx.z; weights pre-padded
  auto G1 = [&](const float* A, int seg) {
    gemm_wmma_f32<<<dim3(NP_ / 64, M_ / 64, 2), 256, 0, stream>>>(
        A, A, KP_, WpA, WpB, NP_, hcat_mu + seg, hcat_lv + seg, HK_,
        HID_, NP_, nullptr, nullptr, 1);
  };

  for (int X = 0; X < 2; ++X) {
    // x = edges (relu'd), then inception hops l1, l2 (l2 reuses xbuf)
    k_edges<<<dim3(40, E_, B_), 256, 0, stream>>>(inciX[X], ops, embWgX[X], xbuf);
    k_wpad<<<dim3(gWpad), 256, 0, stream>>>(Wmu_[X][0], WpA);
    k_wpad<<<dim3(gWpad), 256, 0, stream>>>(Wlv_[X][0], WpB);
    G1(xbuf, 0);
    k_vx<<<dim3(40, V_, B_), 256, 0, stream>>>(inciX[X], xbuf, vxbuf);
    k_lx<<<dim3(40, E_, B_), 256, 0, stream>>>(inciX[X], vxbuf, l1buf);
    k_wpad<<<dim3(gWpad), 256, 0, stream>>>(Wmu_[X][1], WpA);
    k_wpad<<<dim3(gWpad), 256, 0, stream>>>(Wlv_[X][1], WpB);
    G1(l1buf, 512);
    k_vx<<<dim3(40, V_, B_), 256, 0, stream>>>(inciX[X], l1buf, vxbuf);
    k_lx<<<dim3(40, E_, B_), 256, 0, stream>>>(inciX[X], vxbuf, xbuf);   // l2
    k_wpad<<<dim3(gWpad), 256, 0, stream>>>(Wmu_[X][2], WpA);
    k_wpad<<<dim3(gWpad), 256, 0, stream>>>(Wlv_[X][2], WpB);
    G1(xbuf, 1024);

    // [h0|h1|h2] @ Wout + bout  (padded 1536x64 repack of Wout)
    k_woutp<<<dim3((HK_ * 64 + 255) / 256), 256, 0, stream>>>(WoutMu[X], WoM);
    k_woutp<<<dim3((HK_ * 64 + 255) / 256), 256, 0, stream>>>(WoutLv[X], WoL);
    gemm_wmma_f32<<<dim3(1, M_ / 64, 2), 256, 0, stream>>>(
        hcat_mu, hcat_lv, HK_, WoM, WoL, 64, outMu, outLv, Z_,
        Z_, Z_, boutMu[X], boutLv[X], 0);

    k_edge_out<<<dim3(B_), 256, 0, stream>>>(outMu, outLv, gme[X], bme[X],
        gle[X], ble[X], noiseX[X], out, rowbase[X]);
  }
}